// Post_process_deconv_30142080484095
// MI455X (gfx1250) — compile-verified
//
#include <hip/hip_runtime.h>
#include <stdint.h>

// Problem constants (from reference setup_inputs)
#define BATCH   4
#define IMG_H   352
#define IMG_W   1216
#define K2      9

// Tiling: 64x16 outputs per 256-thread block (8 waves on CDNA5 wave32).
// Halo of 6 covers |offset| up to ~5 (randn => ~always); rare misses take a
// zero-padded global fallback path.
#define TILE_W  64
#define TILE_H  16
#define HALO    6
#define LDS_W   (TILE_W + 2 * HALO)   // 76
#define LDS_H   (TILE_H + 2 * HALO)   // 28
#define NTHR    256

__device__ __forceinline__ float fetch_zpad(const float* __restrict__ p, int y, int x) {
  if (y < 0 || y >= IMG_H || x < 0 || x >= IMG_W) return 0.0f;
  return p[(size_t)y * IMG_W + x];
}

__global__ __launch_bounds__(NTHR) void dkn_dcn_kernel(
    const float* __restrict__ depth,    // [B,1,H,W]   hot: keep temporal
    const float* __restrict__ weight,   // [B,9,H,W]   stream-once: non-temporal
    const float* __restrict__ offset,   // [B,18,H,W]  stream-once: non-temporal
    const float* __restrict__ wker9,    // [9] conv kernel (all ones in reference)
    const float* __restrict__ bias1,    // [1]
    float* __restrict__ out)            // [B,1,H,W]   write-once: non-temporal
{
  __shared__ float tile[LDS_H * LDS_W];

  const int tx0 = blockIdx.x * TILE_W;
  const int ty0 = blockIdx.y * TILE_H;
  const int b   = blockIdx.z;
  const int tid = threadIdx.x;

  const size_t plane  = (size_t)IMG_H * IMG_W;
  const float* dplane = depth  + (size_t)b * plane;
  const float* wbase  = weight + (size_t)b * K2 * plane;
  const float* obase  = offset + (size_t)b * 2 * K2 * plane;
  float*       obuf   = out    + (size_t)b * plane;

  // ---- Stage depth tile (+halo, zero-padded) into LDS via async data mover ----
  for (int e = tid; e < LDS_H * LDS_W; e += NTHR) {
    const int r  = e / LDS_W;
    const int c  = e - r * LDS_W;
    const int gy = ty0 - HALO + r;
    const int gx = tx0 - HALO + c;
    if (gy >= 0 && gy < IMG_H && gx >= 0 && gx < IMG_W) {
      // Low 32 bits of a generic shared pointer == wave-relative LDS byte offset.
      unsigned lds_off = (unsigned)(uintptr_t)(&tile[e]);
      unsigned long long gaddr =
          (unsigned long long)(uintptr_t)(dplane + (size_t)gy * IMG_W + gx);
      asm volatile("global_load_async_to_lds_b32 %0, %1, off"
                   :: "v"(lds_off), "v"(gaddr) : "memory");
    } else {
      tile[e] = 0.0f;   // zero padding outside the image
    }
  }
  asm volatile("s_wait_asynccnt 0x0" ::: "memory");
  __syncthreads();

  float wk9[K2];
#pragma unroll
  for (int k = 0; k < K2; ++k) wk9[k] = wker9[k];
  const float b0 = bias1[0];

  const int lx = tid & (TILE_W - 1);   // 0..63 (coalesced along W)
  const int ly = tid >> 6;             // 0..3

  for (int j = 0; j < TILE_H / 4; ++j) {
    const int hy = ty0 + ly + j * 4;
    const int wx = tx0 + lx;
    const size_t pix = (size_t)hy * IMG_W + wx;

    if (j + 1 < TILE_H / 4) {
      __builtin_prefetch(wbase + pix + 4 * (size_t)IMG_W, 0, 0);
      __builtin_prefetch(obase + pix + 4 * (size_t)IMG_W, 0, 0);
    }

    // Mean-center the 9 modulation weights (dkn_residual path). NT loads:
    // these planes are touched exactly once per call.
    float wv[K2];
    float s = 0.0f;
#pragma unroll
    for (int k = 0; k < K2; ++k) {
      wv[k] = __builtin_nontemporal_load(wbase + (size_t)k * plane + pix);
      s += wv[k];
    }
    const float wmean = s * (1.0f / 9.0f);

    // Residual depth (from LDS tile) + bias.
    float acc = tile[(ly + j * 4 + HALO) * LDS_W + (lx + HALO)] + b0;

#pragma unroll
    for (int k = 0; k < K2; ++k) {
      const float dy = __builtin_nontemporal_load(obase + (size_t)(2 * k    ) * plane + pix);
      const float dx = __builtin_nontemporal_load(obase + (size_t)(2 * k + 1) * plane + pix);
      const float yf = (float)(hy - 1 + (k / 3)) + dy;
      const float xf = (float)(wx - 1 + (k % 3)) + dx;
      const float y0f = floorf(yf);
      const float x0f = floorf(xf);
      const float wy1 = yf - y0f, wx1 = xf - x0f;
      const float wy0 = 1.0f - wy1, wx0 = 1.0f - wx1;
      const int y0 = (int)y0f;
      const int x0 = (int)x0f;

      const int r = y0 - (ty0 - HALO);
      const int c = x0 - (tx0 - HALO);
      float v00, v01, v10, v11;
      if ((unsigned)r < (unsigned)(LDS_H - 1) && (unsigned)c < (unsigned)(LDS_W - 1)) {
        // Fast path: all four bilinear corners inside the zero-filled LDS tile.
        const float* t = &tile[r * LDS_W + c];
        v00 = t[0];
        v01 = t[1];
        v10 = t[LDS_W];
        v11 = t[LDS_W + 1];
      } else {
        // Rare far-offset fallback: zero-padded global reads (exact reference
        // semantics); depth stays regular-temporal on purpose.
        v00 = fetch_zpad(dplane, y0,     x0);
        v01 = fetch_zpad(dplane, y0,     x0 + 1);
        v10 = fetch_zpad(dplane, y0 + 1, x0);
        v11 = fetch_zpad(dplane, y0 + 1, x0 + 1);
      }
      const float samp = (v00 * wx0 + v01 * wx1) * wy0 + (v10 * wx0 + v11 * wx1) * wy1;
      acc += (wv[k] - wmean) * wk9[k] * samp;
    }
    __builtin_nontemporal_store(acc, obuf + pix);
  }
}

extern "C" void kernel_launch(void* const* d_in, const int* in_sizes, int n_in,
                              void* d_out, int out_size, void* d_ws, size_t ws_size,
                              hipStream_t stream) {
  (void)in_sizes; (void)n_in; (void)d_ws; (void)ws_size; (void)out_size;
  const float* depth  = (const float*)d_in[0];
  const float* weight = (const float*)d_in[1];
  const float* offset = (const float*)d_in[2];
  const float* wker   = (const float*)d_in[3];
  const float* bias   = (const float*)d_in[4];
  float* out = (float*)d_out;

  dim3 grid(IMG_W / TILE_W, IMG_H / TILE_H, BATCH);  // 19 x 22 x 4 (exact)
  dkn_dcn_kernel<<<grid, NTHR, 0, stream>>>(depth, weight, offset, wker, bias, out);
}